// AttentionSequencePoolingLayer_85538568667409
// MI455X (gfx1250) — compile-verified
//
#include <hip/hip_runtime.h>

typedef __attribute__((ext_vector_type(2))) float v2f;
typedef __attribute__((ext_vector_type(4))) float v4f;
typedef __attribute__((ext_vector_type(8))) float v8f;

#define TQ 200
#define DQ 64
#define TPAD 208
#define KSTRIDE 68            // 68 floats = 272B row pitch: 16B aligned, conflict-free A reads
#define NEGINF (-4294967295.0f)

__launch_bounds__(128, 1)
__global__ void attn_pool_kernel(const float* __restrict__ q,
                                 const float* __restrict__ keys,
                                 const int*   __restrict__ mask,
                                 const float* __restrict__ W1,
                                 const float* __restrict__ b1,
                                 const float* __restrict__ W2,
                                 const float* __restrict__ b2,
                                 const float* __restrict__ W3,
                                 const float* __restrict__ b3,
                                 float* __restrict__ out)
{
    __shared__ float Ks[TPAD * KSTRIDE];     // 56,576 B : keys tile (rows 200..207 zeroed)
    __shared__ float WeffPf[32 * 16 * 2];    //  4,096 B : Weff as (k,k+1) pairs, N padded to 16
    __shared__ float pre1[4][16][8];         //  2,048 B : per-wave h1 pre-activations
    __shared__ float scores_s[TPAD];         //    832 B
    __shared__ float red_s[128];             //    512 B
    __shared__ v2f   out_part[4][32];        //  1,024 B
    __shared__ float W2_s[32], b2_s[4], W3_s[4], cvec[16];
    __shared__ float b3_s;

    const int tid  = threadIdx.x;
    const int b    = blockIdx.x;
    const int lane = tid & 31;
    const int wave = tid >> 5;

    // ---------------- Phase A: stage everything into LDS ----------------
    if (tid < 32) W2_s[tid] = W2[tid];
    if (tid < 4)  { b2_s[tid] = b2[tid]; W3_s[tid] = W3[tid]; }
    if (tid == 0) b3_s = b3[0];

    // keys[b] -> LDS via CDNA5 async DMA (no VGPR round-trip, tracked by ASYNCcnt).
    // LDS byte address for the async op = low 32 bits of the flat __shared__ pointer
    // (flat->LDS truncation per ISA 10.2).
    {
        const float* gk = keys + (size_t)b * (TQ * DQ);
        for (int e = tid; e < TQ * 16; e += 128) {
            int row = e >> 4, c4 = e & 15;
            unsigned ldsAddr = (unsigned)(uintptr_t)&Ks[row * KSTRIDE + c4 * 4];
            const float* ga = gk + e * 4;
            asm volatile("global_load_async_to_lds_b128 %0, %1, off"
                         :: "v"(ldsAddr), "v"(ga)
                         : "memory");
        }
    }

    // zero the 8 padding rows (cols 0..63 are what WMMA reads) - disjoint from async writes
    for (int e = tid; e < 8 * 64; e += 128) {
        int row = TQ + (e >> 6), col = e & 63;
        Ks[row * KSTRIDE + col] = 0.0f;
    }

    // Effective weight: Weff[d][n] = W1k - W1diff + q_d * W1prod (overlaps the async DMA;
    // W1 is 8KB and L2-resident across all 4096 blocks)
    if (tid < 64) {
        int d = tid;
        float qd = q[(size_t)b * DQ + d];
#pragma unroll
        for (int n = 0; n < 8; ++n) {
            float w = W1[(64 + d) * 8 + n] - W1[(128 + d) * 8 + n] + qd * W1[(192 + d) * 8 + n];
            WeffPf[(((d >> 1) * 16 + n) << 1) + (d & 1)] = w;
        }
#pragma unroll
        for (int n = 8; n < 16; ++n)
            WeffPf[(((d >> 1) * 16 + n) << 1) + (d & 1)] = 0.0f;
    }
    // Constant part: c[n] = b1[n] + sum_d q_d * (W1q[d][n] + W1diff[d][n])
    if (tid < 16) {
        float cn = 0.0f;
        if (tid < 8) {
            cn = b1[tid];
            for (int d = 0; d < 64; ++d) {
                float qd = q[(size_t)b * DQ + d];
                cn += qd * (W1[d * 8 + tid] + W1[(128 + d) * 8 + tid]);
            }
        }
        cvec[tid] = cn;
    }

    // all async key DMAs must land before the compute phase
    asm volatile("s_wait_asynccnt 0x0" ::: "memory");
    __syncthreads();

    // ---------------- Phase B: WMMA score GEMM + tiny MLP tail ----------------
    {
        const int m  = lane & 15;      // A: row-in-tile / B: output neuron / C: column
        const int hi = lane >> 4;
        const float cn = cvec[m];
        for (int tile = wave; tile < 13; tile += 4) {
            const int r = tile * 16 + m;
            v8f acc = {cn, cn, cn, cn, cn, cn, cn, cn};
#pragma unroll
            for (int k0 = 0; k0 < 64; k0 += 4) {
                const int ka = k0 + (hi << 1);
                v2f a  = *(const v2f*)&Ks[r * KSTRIDE + ka];
                v2f bb = *(const v2f*)&WeffPf[((((k0 >> 1) + hi) * 16 + m) << 1)];
                acc = __builtin_amdgcn_wmma_f32_16x16x4_f32(false, a, false, bb,
                                                            (short)0, acc, false, false);
            }
            // transpose h1 pre-activations through LDS (lanes with neuron n<8 hold data)
            if (m < 8) {
#pragma unroll
                for (int v = 0; v < 8; ++v) pre1[wave][v + (hi << 3)][m] = acc[v];
            }
            __builtin_amdgcn_fence(__ATOMIC_ACQ_REL, "workgroup");
            if (lane < 16) {
                const int t = tile * 16 + lane;
                float h1[8];
#pragma unroll
                for (int i = 0; i < 8; ++i)
                    h1[i] = 1.0f / (1.0f + __expf(-pre1[wave][lane][i]));
                float s = b3_s;
#pragma unroll
                for (int j = 0; j < 4; ++j) {
                    float a2 = b2_s[j];
#pragma unroll
                    for (int i = 0; i < 8; ++i) a2 += h1[i] * W2_s[i * 4 + j];
                    s += W3_s[j] * (1.0f / (1.0f + __expf(-a2)));
                }
                if (t < TQ) {
                    bool valid = mask[(size_t)b * TQ + t] != 0;
                    scores_s[t] = valid ? s : NEGINF;
                }
            }
            __builtin_amdgcn_fence(__ATOMIC_ACQ_REL, "workgroup");
        }
    }
    __syncthreads();

    // ---------------- Phase C: masked softmax over T=200 ----------------
    float mv = scores_s[tid];
    if (tid < TQ - 128) mv = fmaxf(mv, scores_s[128 + tid]);
    red_s[tid] = mv;
    __syncthreads();
    for (int s = 64; s > 0; s >>= 1) {
        if (tid < s) red_s[tid] = fmaxf(red_s[tid], red_s[tid + s]);
        __syncthreads();
    }
    const float maxv = red_s[0];
    __syncthreads();

    float e0 = __expf(scores_s[tid] - maxv);
    scores_s[tid] = e0;
    float e1 = 0.0f;
    if (tid < TQ - 128) { e1 = __expf(scores_s[128 + tid] - maxv); scores_s[128 + tid] = e1; }
    red_s[tid] = e0 + e1;
    __syncthreads();
    for (int s = 64; s > 0; s >>= 1) {
        if (tid < s) red_s[tid] += red_s[tid + s];
        __syncthreads();
    }
    const float inv = 1.0f / red_s[0];

    // ---------------- Phase D: out = attn @ keys (keys reused from LDS) ----------------
    const int pair = tid & 31;    // 2 output dims per lane
    const int part = tid >> 5;    // 4-way split over T
    v2f acc2 = {0.0f, 0.0f};
    for (int t = part; t < TQ; t += 4) {
        float w  = scores_s[t];
        v2f  kv = *(const v2f*)&Ks[t * KSTRIDE + pair * 2];
        acc2 += w * kv;
    }
    out_part[part][pair] = acc2;
    __syncthreads();
    if (tid < 32) {
        v2f r = out_part[0][tid] + out_part[1][tid] + out_part[2][tid] + out_part[3][tid];
        r *= inv;
        *(v2f*)&out[(size_t)b * DQ + tid * 2] = r;
    }
}

extern "C" void kernel_launch(void* const* d_in, const int* in_sizes, int n_in,
                              void* d_out, int out_size, void* d_ws, size_t ws_size,
                              hipStream_t stream) {
    (void)in_sizes; (void)n_in; (void)out_size; (void)d_ws; (void)ws_size;
    const float* q    = (const float*)d_in[0];
    const float* keys = (const float*)d_in[1];
    const int*   mask = (const int*)d_in[2];
    const float* W1   = (const float*)d_in[3];
    const float* b1   = (const float*)d_in[4];
    const float* W2   = (const float*)d_in[5];
    const float* b2   = (const float*)d_in[6];
    const float* W3   = (const float*)d_in[7];
    const float* b3   = (const float*)d_in[8];
    float* out = (float*)d_out;

    attn_pool_kernel<<<dim3(4096), dim3(128), 0, stream>>>(
        q, keys, mask, W1, b1, W2, b2, W3, b3, out);
}